// GraphSAGEEncoder_70806830841996
// MI455X (gfx1250) — compile-verified
//
#include <hip/hip_runtime.h>

// GraphSAGE 2-layer encoder for MI455X (gfx1250, wave32).
// Roofline: edge aggregation (L2 gathers + float atomics) dominates; GEMMs are
// ~10 GFLOP total -> use exact-precision V_WMMA_F32_16X16X4_F32 for them.

#define N_NODES 100000
#define N_EDGES 1600000
#define IN_DIM  64
#define HID     128

typedef __attribute__((ext_vector_type(2))) float v2f;
typedef __attribute__((ext_vector_type(8))) float v8f;

// ---------------------------------------------------------------- degree ----
__global__ void degree_kernel(const int* __restrict__ dst,
                              float* __restrict__ cnt, int E) {
    int e = blockIdx.x * blockDim.x + threadIdx.x;
    if (e < E) atomicAdd(&cnt[dst[e]], 1.0f);
}

// ------------------------------------------------------------- aggregate ----
// One thread per (edge, feature). Within a wave all 32 lanes share one edge
// (D is 64 or 128), so the gather and the scatter-atomics are both fully
// coalesced 128B transactions that resolve in L2 (feat/h1 fit in 192MB L2).
template <int D>
__global__ void aggregate_kernel(const float* __restrict__ h,
                                 const int* __restrict__ src,
                                 const int* __restrict__ dst,
                                 float* __restrict__ accum, int E) {
    int idx = blockIdx.x * blockDim.x + threadIdx.x;
    if (idx >= E * D) return;
    int e = idx / D;            // D is a power of two -> shift
    int f = idx - e * D;
    float v = h[src[e] * D + f];
    atomicAdd(&accum[dst[e] * D + f], v);
}

// ------------------------------------------------------------ SAGE layer ----
// Block = 16 nodes x 128 output features. 8 waves; wave w owns output columns
// [16w, 16w+16). Per wave: D(16x16,f32) accumulated over K in steps of 4 with
// two v_wmma_f32_16x16x4_f32 per step (self-GEMM + neighbor-GEMM).
//
// Fragment layouts (CDNA5 ISA 7.12.2, 32-bit, wave32):
//   A 16x4 : lane L<16  -> row M=L,    VGPR0=K+0, VGPR1=K+1
//            lane L>=16 -> row M=L-16, VGPR0=K+2, VGPR1=K+3
//   B 4x16 : lane L<16  -> col N=L,    VGPR0=K+0, VGPR1=K+1
//            lane L>=16 -> col N=L-16, VGPR0=K+2, VGPR1=K+3
//   C/D    : VGPR v, lane L -> row v + 8*(L>=16), col L&15
template <int K, bool STORE, bool MEAN>
__global__ __launch_bounds__(256)
void sage_layer_kernel(const float* __restrict__ Hself,   // [N,K] input feats
                       const float* __restrict__ Accum,   // [N,K] neighbor sums
                       const float* __restrict__ Cnt,     // [N]   in-degrees
                       const float* __restrict__ Wself,   // [HID,K]
                       const float* __restrict__ Wneigh,  // [HID,K]
                       const float* __restrict__ Bias,    // [HID]
                       float* __restrict__ Hout,          // [N,HID] (STORE)
                       float* __restrict__ MeanOut,       // [HID]   (MEAN)
                       float invN) {
    constexpr int LDK = K + 2;             // even stride, conflict-free banks
    __shared__ float sSelf[16 * LDK];
    __shared__ float sNeigh[16 * LDK];
    __shared__ float sInv[16];
    __shared__ float sOut[HID];

    const int tid = threadIdx.x;
    const int node_base = blockIdx.x * 16;

    if (tid < 16) {
        float c = Cnt[node_base + tid];
        sInv[tid] = (c > 0.0f) ? (1.0f / c) : 0.0f;   // zero-indeg -> 0 mean
    }
    if (MEAN && tid < HID) sOut[tid] = 0.0f;
    __syncthreads();

    // Stage the 16-node A panels once per block (reused by all 8 waves);
    // fold the 1/deg of the segment-mean into the neighbor panel here.
    for (int i = tid; i < 16 * K; i += 256) {
        int m = i / K;
        int k = i - m * K;
        int node = node_base + m;
        sSelf[m * LDK + k]  = Hself[node * K + k];
        sNeigh[m * LDK + k] = Accum[node * K + k] * sInv[m];
    }
    __syncthreads();

    const int wave  = tid >> 5;
    const int lane  = tid & 31;
    const int n     = lane & 15;     // tile column
    const int m     = lane & 15;     // tile row (A side)
    const int hi    = lane >> 4;     // 0: K+{0,1}, 1: K+{2,3}
    const int jbase = wave * 16;
    const int j     = jbase + n;

    const float* __restrict__ wrow_s = Wself  + j * K;   // L2-broadcast hits
    const float* __restrict__ wrow_n = Wneigh + j * K;
    const float* as_base = sSelf  + m * LDK;
    const float* an_base = sNeigh + m * LDK;

    v8f c = {0.f, 0.f, 0.f, 0.f, 0.f, 0.f, 0.f, 0.f};
#pragma unroll
    for (int k = 0; k < K; k += 4) {
        const int kk = k + 2 * hi;                 // even -> 8B aligned
        v2f a_s = *(const v2f*)(as_base + kk);     // ds_load_b64
        v2f b_s = *(const v2f*)(wrow_s + kk);      // global_load_b64
        c = __builtin_amdgcn_wmma_f32_16x16x4_f32(false, a_s, false, b_s,
                                                  (short)0, c, false, false);
        v2f a_n = *(const v2f*)(an_base + kk);
        v2f b_n = *(const v2f*)(wrow_n + kk);
        c = __builtin_amdgcn_wmma_f32_16x16x4_f32(false, a_n, false, b_n,
                                                  (short)0, c, false, false);
    }

    // Epilogue: bias + ReLU, then store h1 and/or accumulate the graph mean.
    const float bj = Bias[j];
    float s = 0.0f;
#pragma unroll
    for (int v = 0; v < 8; ++v) {
        int row = v + 8 * hi;
        float val = c[v] + bj;
        val = fmaxf(val, 0.0f);
        if (STORE) Hout[(size_t)(node_base + row) * HID + j] = val;
        if (MEAN)  s += val;
    }
    if (MEAN) {
        atomicAdd(&sOut[j], s);                    // ds_add_f32 (lanes L,L+16)
        __syncthreads();
        if (tid < HID) atomicAdd(&MeanOut[tid], sOut[tid] * invN);
    }
}

// ---------------------------------------------------------------- launch ----
extern "C" void kernel_launch(void* const* d_in, const int* in_sizes, int n_in,
                              void* d_out, int out_size, void* d_ws, size_t ws_size,
                              hipStream_t stream) {
    const float* feat    = (const float*)d_in[0];
    const int*   src     = (const int*)  d_in[1];
    const int*   dst     = (const int*)  d_in[2];
    const float* Wself1  = (const float*)d_in[3];
    const float* b1      = (const float*)d_in[4];
    const float* Wneigh1 = (const float*)d_in[5];
    const float* Wself2  = (const float*)d_in[6];
    const float* b2      = (const float*)d_in[7];
    const float* Wneigh2 = (const float*)d_in[8];
    float* out = (float*)d_out;

    // Workspace layout (floats):
    //   accum1 [N*64] | accum2 [N*128] | cnt [N] | h1 [N*128]   (~129 MB)
    float* ws     = (float*)d_ws;
    float* accum1 = ws;
    float* accum2 = accum1 + (size_t)N_NODES * IN_DIM;
    float* cnt    = accum2 + (size_t)N_NODES * HID;
    float* h1     = cnt + N_NODES;

    // Zero accumulators + degrees (one contiguous region) and the output.
    size_t zero_bytes =
        ((size_t)N_NODES * IN_DIM + (size_t)N_NODES * HID + N_NODES) * sizeof(float);
    hipMemsetAsync(accum1, 0, zero_bytes, stream);
    hipMemsetAsync(out, 0, (size_t)out_size * sizeof(float), stream);

    // In-degrees (shared by both layers).
    degree_kernel<<<(N_EDGES + 255) / 256, 256, 0, stream>>>(dst, cnt, N_EDGES);

    // Layer 1: aggregate feat -> accum1, then GEMM+ReLU -> h1.
    {
        int total = N_EDGES * IN_DIM;
        aggregate_kernel<IN_DIM><<<(total + 255) / 256, 256, 0, stream>>>(
            feat, src, dst, accum1, N_EDGES);
        sage_layer_kernel<IN_DIM, true, false><<<N_NODES / 16, 256, 0, stream>>>(
            feat, accum1, cnt, Wself1, b1, Wneigh1, h1, nullptr, 0.0f);
    }

    // Layer 2: aggregate h1 -> accum2, then GEMM+ReLU fused with graph-mean
    // (h2 is never materialized).
    {
        int total = N_EDGES * HID;
        aggregate_kernel<HID><<<(total + 255) / 256, 256, 0, stream>>>(
            h1, src, dst, accum2, N_EDGES);
        sage_layer_kernel<HID, false, true><<<N_NODES / 16, 256, 0, stream>>>(
            h1, accum2, cnt, Wself2, b2, Wneigh2, nullptr, out,
            1.0f / (float)N_NODES);
    }
}